// GCN_56092272885944
// MI455X (gfx1250) — compile-verified
//
#include <hip/hip_runtime.h>

#define NN 100000      // N_NODES
#define NE 1600000     // N_EDGES
#define NG 1000        // N_GRAPHS
#define EMB 64
#define WP 80          // weight LDS pitch in K-pairs (lo/hi halves hit disjoint bank sets)
#define TSTR 68        // activation staging pitch (aligned v2f frag loads, conflict-free)
#define MAXB 512       // persistent-block cap for the node-MLP grid

typedef float v2f __attribute__((ext_vector_type(2)));
typedef float v8f __attribute__((ext_vector_type(8)));

// ---------------- agg = (1 + eps[conv]) * h  (flat over N*EMB) ----------------
__global__ __launch_bounds__(256) void k_scale(const float* __restrict__ hin,
                                               const float* __restrict__ eps, int conv,
                                               float* __restrict__ agg, int n) {
  int i = blockIdx.x * 256 + threadIdx.x;
  if (i < n) agg[i] = (1.0f + eps[conv]) * hin[i];
}

// ---------------- scatter: agg[dst] += h[src]; one float4 quad per thread ----------------
__global__ __launch_bounds__(256) void k_scatter(const float* __restrict__ hin,
                                                 const int* __restrict__ src,
                                                 const int* __restrict__ dst,
                                                 float* __restrict__ agg) {
  long long i = (long long)blockIdx.x * 256 + threadIdx.x;  // (edge, quad-of-4-floats)
  if (i >= (long long)NE * (EMB / 4)) return;
  int e = (int)(i >> 4);
  int q = ((int)i & 15) * 4;
  int s = src[e], d = dst[e];
  float4 v = *(const float4*)(hin + (size_t)s * EMB + q);
  float* p = agg + (size_t)d * EMB + q;
  atomicAdd(p + 0, v.x);
  atomicAdd(p + 1, v.y);
  atomicAdd(p + 2, v.z);
  atomicAdd(p + 3, v.w);
}

// ---------------- fused 2-layer MLP via V_WMMA_F32_16X16X4_F32, persistent blocks --------
// out = L2( relu(in @ W1 + B1) ),  L2(y) = y @ W2 + B2 (+ReLU if relu2).
// One wave32 per 16-row tile, 4 waves/block, grid-stride over tiles.
__global__ __launch_bounds__(128) void k_mlp(const float* __restrict__ in,
                                             const float* __restrict__ W1,
                                             const float* __restrict__ B1,
                                             const float* __restrict__ W2,
                                             const float* __restrict__ B2,
                                             float* __restrict__ out,
                                             int nrows, int tiles, int relu2) {
  // Paired-K weight layout: w1p[p*WP + n] = (W[2p][n], W[2p+1][n]); p = 0..31.
  __shared__ v2f   w1p[32 * WP];
  __shared__ v2f   w2p[32 * WP];
  __shared__ float b1s[EMB], b2s[EMB];
  __shared__ float stage[4][16 * TSTR];

  const int tid = threadIdx.x;
  for (int i = tid; i < 32 * EMB; i += 128) {
    int p = i >> 6, n = i & 63;
    v2f a1 = {W1[(2 * p) * EMB + n], W1[(2 * p + 1) * EMB + n]};
    v2f a2 = {W2[(2 * p) * EMB + n], W2[(2 * p + 1) * EMB + n]};
    w1p[p * WP + n] = a1;
    w2p[p * WP + n] = a2;
  }
  if (tid < EMB) { b1s[tid] = B1[tid]; b2s[tid] = B2[tid]; }
  __syncthreads();

  const int wave = tid >> 5, lane = tid & 31;
  const int lcol = lane & 15;   // N index (or M for A-frags) within tile
  const int lhi  = lane >> 4;   // half-wave selector
  float* st = stage[wave];

  for (int t0 = blockIdx.x * 4; t0 < tiles; t0 += gridDim.x * 4) {
    const int row0 = (t0 + wave) * 16;

    // Stage the 16xEMB input tile (coalesced float4 global reads -> b128 LDS stores).
    for (int t = lane; t < 16 * 16; t += 32) {
      int r = t >> 4, c4 = (t & 15) * 4;
      int gr = row0 + r;
      float4 v = (gr < nrows) ? *(const float4*)(in + (size_t)gr * EMB + c4)
                              : make_float4(0.f, 0.f, 0.f, 0.f);
      *(float4*)(st + r * TSTR + c4) = v;
    }
    __syncthreads();

    // A fragments, layer 1: aligned v2f loads, A[v][lane] = Z[M=lcol][K=4kt+v+2lhi].
    v2f a[16];
#pragma unroll
    for (int kt = 0; kt < 16; ++kt)
      a[kt] = *(const v2f*)(st + lcol * TSTR + kt * 4 + 2 * lhi);

    // Layer 1: H1 = relu(Z @ W1 + b1), restaged (C-layout -> A-layout).
#pragma unroll
    for (int nt = 0; nt < 4; ++nt) {
      v8f acc = {0.f, 0.f, 0.f, 0.f, 0.f, 0.f, 0.f, 0.f};
#pragma unroll
      for (int kt = 0; kt < 16; ++kt) {
        v2f bv = w1p[(kt * 2 + lhi) * WP + nt * 16 + lcol];
        acc = __builtin_amdgcn_wmma_f32_16x16x4_f32(false, a[kt], false, bv,
                                                    (short)0, acc, false, false);
      }
      float bias = b1s[nt * 16 + lcol];
#pragma unroll
      for (int v = 0; v < 8; ++v) {
        float x = acc[v] + bias;                    // C-layout: row = v + 8*lhi
        st[(v + 8 * lhi) * TSTR + nt * 16 + lcol] = x > 0.f ? x : 0.f;
      }
    }
    __syncthreads();

    // A fragments, layer 2.
#pragma unroll
    for (int kt = 0; kt < 16; ++kt)
      a[kt] = *(const v2f*)(st + lcol * TSTR + kt * 4 + 2 * lhi);

    // Layer 2: out = H1 @ W2 + b2 (ReLU only for the head), restaged for coalesced stores.
#pragma unroll
    for (int nt = 0; nt < 4; ++nt) {
      v8f acc = {0.f, 0.f, 0.f, 0.f, 0.f, 0.f, 0.f, 0.f};
#pragma unroll
      for (int kt = 0; kt < 16; ++kt) {
        v2f bv = w2p[(kt * 2 + lhi) * WP + nt * 16 + lcol];
        acc = __builtin_amdgcn_wmma_f32_16x16x4_f32(false, a[kt], false, bv,
                                                    (short)0, acc, false, false);
      }
      float bias = b2s[nt * 16 + lcol];
#pragma unroll
      for (int v = 0; v < 8; ++v) {
        float x = acc[v] + bias;
        if (relu2) x = x > 0.f ? x : 0.f;
        st[(v + 8 * lhi) * TSTR + nt * 16 + lcol] = x;
      }
    }
    __syncthreads();

    // Coalesced float4 stores of the finished tile.
    for (int t = lane; t < 16 * 16; t += 32) {
      int r = t >> 4, c4 = (t & 15) * 4;
      int gr = row0 + r;
      if (gr < nrows)
        *(float4*)(out + (size_t)gr * EMB + c4) = *(const float4*)(st + r * TSTR + c4);
    }
    __syncthreads();   // protect staging reuse across loop iterations
  }
}

// ---------------- zero a flat float buffer ----------------
__global__ __launch_bounds__(256) void k_zero(float* __restrict__ p, int n) {
  int i = blockIdx.x * 256 + threadIdx.x;
  if (i < n) p[i] = 0.0f;
}

// ---------------- segment-sum pooling: sums[batch[n]] += h[n]; counts ----------------
__global__ __launch_bounds__(256) void k_pool(const float* __restrict__ h,
                                              const int* __restrict__ batch,
                                              float* __restrict__ sums,
                                              float* __restrict__ cnts, int n) {
  int i = blockIdx.x * 256 + threadIdx.x;
  if (i >= n) return;
  int node = i >> 6, f = i & 63;
  int g = batch[node];
  atomicAdd(&sums[(size_t)g * EMB + f], h[i]);
  if (f == 0) atomicAdd(&cnts[g], 1.0f);
}

// ---------------- pooled = sums / max(counts, 1) ----------------
__global__ __launch_bounds__(256) void k_div(const float* __restrict__ sums,
                                             const float* __restrict__ cnts,
                                             float* __restrict__ pooled, int n) {
  int i = blockIdx.x * 256 + threadIdx.x;
  if (i >= n) return;
  int g = i >> 6;
  pooled[i] = sums[i] / fmaxf(cnts[g], 1.0f);
}

extern "C" void kernel_launch(void* const* d_in, const int* in_sizes, int n_in,
                              void* d_out, int out_size, void* d_ws, size_t ws_size,
                              hipStream_t stream) {
  (void)in_sizes; (void)n_in; (void)out_size; (void)ws_size;
  const float* x     = (const float*)d_in[0];
  const int*   ei    = (const int*)d_in[1];     // [2, NE]: src row then dst row
  /* d_in[2] = edge_attr (unused by GIN reference) */
  const int*   batch = (const int*)d_in[3];
  const float* w1    = (const float*)d_in[4];
  const float* b1    = (const float*)d_in[5];
  const float* w2    = (const float*)d_in[6];
  const float* b2    = (const float*)d_in[7];
  const float* eps   = (const float*)d_in[8];
  const float* lw1   = (const float*)d_in[9];
  const float* lb1   = (const float*)d_in[10];
  const float* lw2   = (const float*)d_in[11];
  const float* lb2   = (const float*)d_in[12];

  float* agg    = (float*)d_ws;                       // N*EMB
  float* h      = agg  + (size_t)NN * EMB;            // N*EMB
  float* sums   = h    + (size_t)NN * EMB;            // NG*EMB
  float* cnts   = sums + (size_t)NG * EMB;            // NG
  float* pooled = cnts + NG;                          // NG*EMB

  const int elemsN   = NN * EMB;                      // 6.4M
  const int gScale   = (elemsN + 255) / 256;
  const int gScatter = (int)(((long long)NE * 16 + 255) / 256);
  const int tilesN   = (NN + 15) / 16;
  int gMlpN          = (tilesN + 3) / 4;
  if (gMlpN > MAXB) gMlpN = MAXB;                     // persistent blocks

  for (int c = 0; c < 3; ++c) {
    const float* hin = c ? h : x;
    // agg := (1+eps)*h, then scatter-add neighbors on top -> agg == z
    k_scale  <<<gScale,   256, 0, stream>>>(hin, eps, c, agg, elemsN);
    k_scatter<<<gScatter, 256, 0, stream>>>(hin, ei, ei + NE, agg);
    k_mlp    <<<gMlpN,    128, 0, stream>>>(agg, w1 + c * EMB * EMB, b1 + c * EMB,
                                            w2 + c * EMB * EMB, b2 + c * EMB,
                                            h, NN, tilesN, /*relu2=*/0);
  }

  const int zN = NG * EMB + NG;                       // sums and cnts are contiguous
  k_zero<<<(zN + 255) / 256, 256, 0, stream>>>(sums, zN);
  k_pool<<<gScale, 256, 0, stream>>>(h, batch, sums, cnts, elemsN);
  k_div <<<(NG * EMB + 255) / 256, 256, 0, stream>>>(sums, cnts, pooled, NG * EMB);

  const int tilesG = (NG + 15) / 16;
  const int gMlpG  = (tilesG + 3) / 4;
  k_mlp<<<gMlpG, 128, 0, stream>>>(pooled, lw1, lb1, lw2, lb2,
                                   (float*)d_out, NG, tilesG, /*relu2=*/1);
}